// CausalADGLoss_30245159698453
// MI455X (gfx1250) — compile-verified
//
#include <hip/hip_runtime.h>

// CDNA5 / gfx1250, wave32.
typedef __attribute__((ext_vector_type(2))) float v2f;
typedef __attribute__((ext_vector_type(8))) float v8f;

#define B_       32
#define C_       2
#define T_       192000
#define TDS      48000          // downsampled length (::4)
#define TS       75             // steps per LDS tile (divisible by 3 -> uniform loss loop)
#define TSP      77             // padded LDS row stride (odd -> bank-conflict-free)
#define NTILES   (TDS / TS)     // 640 (exact)
#define LSTEPS   (TS / 3)       // 25 loss points per thread per tile (constant)
#define NTHREADS 192            // 3 signals * 64 (b,c) chains = 6 wave32s
#define BSTRIDE  (T_ * C_)      // 384000 floats per batch
#define DSSTRIDE 8              // floats between ::4 samples (4*C)

__global__ __launch_bounds__(NTHREADS)
void CausalADGLoss_kernel(const float* __restrict__ pred,
                          const float* __restrict__ target,
                          const float* __restrict__ input,
                          float* __restrict__ out)
{
    // envT: [signal(3)][bc(64)][TSP] envelope tiles.  59136 B
    __shared__ float  envT[3 * 64 * TSP];
    // red: final f64 partials.                         3072 B  (total 62208 B)
    __shared__ double red[2 * NTHREADS];

    const int tid = threadIdx.x;
    const int sig = tid >> 6;          // 0=input, 1=target, 2=pred (uniform per wave)
    const int bc  = tid & 63;
    const int b   = bc >> 1;
    const int c   = bc & 1;

    const float* xs = (sig == 0) ? input : (sig == 1) ? target : pred;
    const float* gp = xs + (size_t)b * BSTRIDE + c;

    // Gains exactly as reference: float32(exp(-1/(fs*tau)))
    const float GA = 0.99584200184510607f;   // attack  (5 ms)
    const float GR = 0.99930579662628243f;   // release (30 ms)  (GR > GA)
    const float KA = 1.0f - GA;
    const float KR = 1.0f - GR;
    const float EPSF = 1.1920928955078125e-7f;  // float32 eps

    float  st   = 0.0f;        // envelope state for this chain
    double accA = 0.0;         // sum (a-o)^2  (wave-WMMA-reduced, x2 duplicated)
    double accB = 0.0;         // sum o^2
    const v2f onesA = {1.0f, 1.0f};

    float* myRow = &envT[tid * TSP];

    for (int tile = 0; tile < NTILES; ++tile) {
        const float* tp = gp + (size_t)tile * TS * DSSTRIDE;

        // ---- scan phase: 75 serial IIR steps, branch-free max-of-FMA form ----
        // env' = max(fma(ga, env, (1-ga)|x|), fma(gr, env, (1-gr)|x|))
        // (gr > ga  =>  max picks exactly the reference's s>state branch)
        #pragma unroll 5
        for (int i = 0; i < TS; ++i) {
            if (i % 5 == 0)     // static under unroll-5; prefetch ~2 tiles ahead
                __builtin_prefetch(tp + i * DSSTRIDE + 2 * TS * DSSTRIDE, 0, 1);
            float x  = tp[i * DSSTRIDE];
            float s  = __builtin_fabsf(x);           // free |src| modifier
            float f1 = __builtin_fmaf(GA, st, KA * s);
            float f2 = __builtin_fmaf(GR, st, KR * s);
            st = fmaxf(f1, f2);
            myRow[i] = st;
        }
        __syncthreads();

        // ---- loss phase: thread covers its bc at steps sig, sig+3, ...  ----
        // Constant 25-iteration bound -> no divergent-loop EXEC bookkeeping.
        float tA = 0.0f, tB = 0.0f;
        const float* rin = &envT[(0 * 64 + bc) * TSP + sig];
        const float* rtg = &envT[(1 * 64 + bc) * TSP + sig];
        const float* rpr = &envT[(2 * 64 + bc) * TSP + sig];
        #pragma unroll 5
        for (int k = 0; k < LSTEPS; ++k) {
            float ie = rin[3 * k], te = rtg[3 * k], pe = rpr[3 * k];
            float r  = 1.0f / (ie + EPSF);
            float d  = (te - pe) * r;                // a - o
            float o  = pe * r;
            tA = __builtin_fmaf(d, d, tA);
            tB = __builtin_fmaf(o, o, tB);
        }

        // ---- wave-level exact-f32 reduction via V_WMMA_F32_16X16X4_F32 ----
        // A = ones(16x4), B = per-lane (tA, 0): D[m][n] = column sums of B.
        // Sum of D VGPR0 over all 32 lanes = 2 * (wave total); halved at end.
        // EXEC is all-1s here (uniform control flow, full 192-thread block).
        {
            v8f zero8 = {0, 0, 0, 0, 0, 0, 0, 0};
            v2f bA = {tA, 0.0f};
            v2f bB = {tB, 0.0f};
            v8f dA = __builtin_amdgcn_wmma_f32_16x16x4_f32(
                false, onesA, false, bA, (short)0, zero8, false, false);
            v8f dB = __builtin_amdgcn_wmma_f32_16x16x4_f32(
                false, onesA, false, bB, (short)0, zero8, false, false);
            accA += (double)dA[0];
            accB += (double)dB[0];
        }
        __syncthreads();   // envT safe to overwrite next tile
    }

    // ---- final f64 reduction over the 192 lanes (x2 duplication -> *0.5) ----
    red[tid]            = accA;
    red[NTHREADS + tid] = accB;
    __syncthreads();
    if (tid == 0) {
        double SA = 0.0, SB = 0.0;
        for (int i = 0; i < NTHREADS; ++i) {
            SA += red[i];
            SB += red[NTHREADS + i];
        }
        SA *= 0.5;           // WMMA D rows are duplicated (M=0 and M=8 halves)
        SB *= 0.5;
        const double N = 3072000.0;  // 32 * 48000 * 2 ds points (repeat is mean-invariant)
        double mse = SA / N;
        double tns = SB / N;
        out[0] = (float)(mse / (tns + 1.1920928955078125e-7));
    }
}

extern "C" void kernel_launch(void* const* d_in, const int* in_sizes, int n_in,
                              void* d_out, int out_size, void* d_ws, size_t ws_size,
                              hipStream_t stream) {
    (void)in_sizes; (void)n_in; (void)out_size; (void)d_ws; (void)ws_size;
    // setup_inputs() order: pred, target, input  (all f32, 32*192000*2)
    const float* pred   = (const float*)d_in[0];
    const float* target = (const float*)d_in[1];
    const float* input  = (const float*)d_in[2];
    float* out = (float*)d_out;

    // 192 chains is the total available parallelism for this sequential IIR:
    // one block of 6 wave32s, everything fused (no workspace traffic).
    CausalADGLoss_kernel<<<1, NTHREADS, 0, stream>>>(pred, target, input, out);
}